// FCGAT_91250875171023
// MI455X (gfx1250) — compile-verified
//
#include <hip/hip_runtime.h>

#define N_    64
#define K_    512
#define D_    256
#define ROWS  (N_ * K_)      // 32768
#define SLOPE 0.01f

typedef float v2f __attribute__((ext_vector_type(2)));
typedef float v8f __attribute__((ext_vector_type(8)));

__device__ __forceinline__ float warp_sum(float v) {
#pragma unroll
    for (int m = 16; m > 0; m >>= 1) v += __shfl_xor(v, m, 32);
    return v;
}
__device__ __forceinline__ float warp_max(float v) {
#pragma unroll
    for (int m = 16; m > 0; m >>= 1) v = fmaxf(v, __shfl_xor(v, m, 32));
    return v;
}
__device__ __forceinline__ float lrelu(float t) { return t > 0.0f ? t : SLOPE * t; }

// ---------------------------------------------------------------------------
// Kernel 1: fold W through the attention vector.
//   u[d] = sum_e W[e,d]*aw[e], v[d] = sum_e W[e,d]*aw[D+e]
//   c12[0] = b.aw[:D], c12[1] = b.aw[D:]
// ---------------------------------------------------------------------------
__global__ void prep_kernel(const float* __restrict__ W, const float* __restrict__ Wb,
                            const float* __restrict__ aw,
                            float* __restrict__ u, float* __restrict__ v,
                            float* __restrict__ c12) {
    int d = threadIdx.x;                    // 256 threads, one per d
    float ua = 0.0f, va = 0.0f;
    for (int e = 0; e < D_; ++e) {
        float w = W[e * D_ + d];            // coalesced across threads
        ua = fmaf(w, aw[e], ua);
        va = fmaf(w, aw[D_ + e], va);
    }
    u[d] = ua;
    v[d] = va;
    if (d < 2) {
        const float* a2 = aw + d * D_;
        float c = 0.0f;
        for (int e = 0; e < D_; ++e) c = fmaf(Wb[e], a2[e], c);
        c12[d] = c;
    }
}

// ---------------------------------------------------------------------------
// Kernel 2: per-row attention scores directly from X.
//   s_src[row] = x_row.u + c1 ; s_dst[row] = x_row.v + c2
// One wave per row, 8 rows per 256-thread block.
// ---------------------------------------------------------------------------
__global__ __launch_bounds__(256) void score_kernel(
    const float* __restrict__ X, const float* __restrict__ u,
    const float* __restrict__ v, const float* __restrict__ c12,
    float* __restrict__ ssrc, float* __restrict__ sdst) {
    __shared__ float us[D_], vs[D_];
    int t = threadIdx.x;
    us[t] = u[t];
    vs[t] = v[t];
    __syncthreads();

    int lane = t & 31, wave = t >> 5;
    int row = blockIdx.x * 8 + wave;
    const float* xr = X + (size_t)row * D_;
    float su = 0.0f, sv = 0.0f;
#pragma unroll
    for (int i = 0; i < D_ / 32; ++i) {
        float x = xr[lane + 32 * i];        // coalesced 128B per wave
        su = fmaf(x, us[lane + 32 * i], su);
        sv = fmaf(x, vs[lane + 32 * i], sv);
    }
    su = warp_sum(su);
    sv = warp_sum(sv);
    if (lane == 0) {
        ssrc[row] = su + c12[0];
        sdst[row] = sv + c12[1];
    }
}

// ---------------------------------------------------------------------------
// Kernel 3: scale[n,i] = sum_j softmax_j( lrelu(s_src[i]+s_dst[j]+ab) )
// One wave per (n,i); s_dst row cached in registers (16 vals/lane).
// ---------------------------------------------------------------------------
__global__ __launch_bounds__(256) void softmax_scale_kernel(
    const float* __restrict__ ssrc, const float* __restrict__ sdst,
    const float* __restrict__ ab, float* __restrict__ scale) {
    int t = threadIdx.x;
    int lane = t & 31, wave = t >> 5;
    int row = blockIdx.x * 8 + wave;        // flattened (n,i)
    int n = row / K_;
    float si = ssrc[row];
    float b = ab[0];
    const float* sd = sdst + (size_t)n * K_;

    float sl[K_ / 32];
    float ml = -3.4e38f;
#pragma unroll
    for (int i = 0; i < K_ / 32; ++i) {
        float x = sd[lane + 32 * i];
        sl[i] = x;
        ml = fmaxf(ml, x);
    }
    float m = warp_max(ml);                     // lrelu monotonic -> row max
    float emax = lrelu(si + m + b);

    float S = 0.0f;
#pragma unroll
    for (int i = 0; i < K_ / 32; ++i) S += __expf(lrelu(si + sl[i] + b) - emax);
    S = warp_sum(S);

    float q = 0.0f;
#pragma unroll
    for (int i = 0; i < K_ / 32; ++i) q += __expf(lrelu(si + sl[i] + b) - emax) / S;
    q = warp_sum(q);
    if (lane == 0) scale[row] = q;
}

// ---------------------------------------------------------------------------
// Kernel 4: h = X.W^T + b via V_WMMA_F32_16X16X4_F32, fused epilogue
//   out = relu(scale * h).
// Block = 256 threads = 8 waves -> 128-row x 16-col tile. W-tile in LDS.
// ---------------------------------------------------------------------------
__global__ __launch_bounds__(256) void gat_gemm_kernel(
    const float* __restrict__ X, const float* __restrict__ W,
    const float* __restrict__ Wb, const float* __restrict__ scale,
    float* __restrict__ out) {
    // Bs[p][c] = { W[colbase+c][2p], W[colbase+c][2p+1] }  (p = d/2)
    __shared__ v2f Bs[(D_ / 2) * 16];      // 16 KB

    int t = threadIdx.x;
    int colbase = blockIdx.y * 16;
    {
        int c = t >> 4;                    // 0..15 column of tile
        int l = t & 15;                    // 0..15
        const float* wrow = W + (size_t)(colbase + c) * D_;
#pragma unroll
        for (int it = 0; it < 8; ++it) {
            int p = l + 16 * it;           // 0..127 ; 16 threads -> 128B coalesced
            Bs[p * 16 + c] = *(const v2f*)(wrow + 2 * p);
        }
    }
    __syncthreads();

    int lane = t & 31, wave = t >> 5;
    int rowbase = blockIdx.x * 128 + wave * 16;
    int m  = lane & 15;                    // A row in tile / B,C col in tile
    int kh = (lane >> 4) << 1;             // K sub-offset: 0 or 2

    const float* Ar = X + (size_t)(rowbase + m) * D_ + kh;
    v8f acc = {0.0f, 0.0f, 0.0f, 0.0f, 0.0f, 0.0f, 0.0f, 0.0f};

#pragma unroll 4
    for (int k0 = 0; k0 < D_; k0 += 4) {
        v2f a = *(const v2f*)(Ar + k0);                     // A frag: 2 consec K
        v2f bfrag = Bs[((k0 + kh) >> 1) * 16 + m];          // B frag: ds_read_b64
        acc = __builtin_amdgcn_wmma_f32_16x16x4_f32(
            /*neg_a=*/false, a, /*neg_b=*/false, bfrag,
            /*c_mod=*/(short)0, acc, /*reuse_a=*/false, /*reuse_b=*/false);
    }

    // Epilogue: C/D layout -> row = rowbase + r + 8*(lane/16), col = colbase + m
    float bcol = Wb[colbase + m];
    int rbase = rowbase + ((lane >> 4) << 3);
#pragma unroll
    for (int r = 0; r < 8; ++r) {
        int row = rbase + r;
        float h = acc[r] + bcol;
        float o = scale[row] * h;
        out[(size_t)row * D_ + colbase + m] = fmaxf(o, 0.0f);
    }
}

// ---------------------------------------------------------------------------
extern "C" void kernel_launch(void* const* d_in, const int* in_sizes, int n_in,
                              void* d_out, int out_size, void* d_ws, size_t ws_size,
                              hipStream_t stream) {
    const float* X  = (const float*)d_in[0];   // [N,K,D]
    const float* W  = (const float*)d_in[1];   // [D,D]
    const float* Wb = (const float*)d_in[2];   // [D]
    const float* aw = (const float*)d_in[3];   // [2D]
    const float* ab = (const float*)d_in[4];   // scalar
    float* out = (float*)d_out;

    float* ws    = (float*)d_ws;
    float* u     = ws;                 // 256
    float* v     = ws + D_;            // 256
    float* c12   = ws + 2 * D_;        // 2
    float* ssrc  = ws + 1024;          // 32768
    float* sdst  = ssrc + ROWS;        // 32768
    float* scale = sdst + ROWS;        // 32768

    prep_kernel<<<1, 256, 0, stream>>>(W, Wb, aw, u, v, c12);
    score_kernel<<<ROWS / 8, 256, 0, stream>>>(X, u, v, c12, ssrc, sdst);
    softmax_scale_kernel<<<ROWS / 8, 256, 0, stream>>>(ssrc, sdst, ab, scale);
    gat_gemm_kernel<<<dim3(ROWS / 128, D_ / 16), 256, 0, stream>>>(X, W, Wb, scale, out);
}